// NSAB_8864812499052
// MI455X (gfx1250) — compile-verified
//
#include <hip/hip_runtime.h>
#include <stdint.h>

#define RBLOCKS 1024
#define TPB     256
#define DEPTH   8          // async pipeline depth; s_wait_asynccnt imm below assumes 8
#define NACC    9          // accumulated components per thread

typedef float v2f __attribute__((ext_vector_type(2)));

// ---------------------------------------------------------------------------
// fast tanh: gfx1250 V_TANH_F32 (TRANS op). Prefer the builtin (compiler
// handles the trans co-exec hazard); fall back to inline asm + v_nop.
// ---------------------------------------------------------------------------
__device__ __forceinline__ float fast_tanhf(float x) {
#if __has_builtin(__builtin_amdgcn_tanhf)
    return __builtin_amdgcn_tanhf(x);
#else
    float r;
    asm("v_tanh_f32 %0, %1\n\tv_nop" : "=v"(r) : "v"(x));
    return r;
#endif
}

// ---------------------------------------------------------------------------
// Packed fp32 VOP3P ops (CDNA5: V_PK_FMA_F32 / V_PK_MUL_F32 / V_PK_ADD_F32).
// The backend scalarizes <2 x float> math for gfx1250, so emit them directly.
// Non-volatile pure-register asm => freely schedulable by the compiler.
// ---------------------------------------------------------------------------
__device__ __forceinline__ v2f pk_fma(v2f a, v2f b, v2f c) {
    v2f d;
    asm("v_pk_fma_f32 %0, %1, %2, %3" : "=v"(d) : "v"(a), "v"(b), "v"(c));
    return d;
}
__device__ __forceinline__ v2f pk_fma_nc(v2f a, v2f b, v2f c) {   // a*b - c
    v2f d;
    asm("v_pk_fma_f32 %0, %1, %2, %3 neg_lo:[0,0,1] neg_hi:[0,0,1]"
        : "=v"(d) : "v"(a), "v"(b), "v"(c));
    return d;
}
__device__ __forceinline__ v2f pk_mul(v2f a, v2f b) {
    v2f d;
    asm("v_pk_mul_f32 %0, %1, %2" : "=v"(d) : "v"(a), "v"(b));
    return d;
}
__device__ __forceinline__ void pk_acc_add(v2f& acc, v2f a) {        // acc += a
    asm("v_pk_add_f32 %0, %0, %1" : "+v"(acc) : "v"(a));
}
__device__ __forceinline__ void pk_acc_fma(v2f& acc, v2f a, v2f b) { // acc += a*b
    asm("v_pk_fma_f32 %0, %1, %2, %0" : "+v"(acc) : "v"(a), "v"(b));
}

// Per-PAIR update of 9 running sums (all packed-f32 except v_tanh_f32).
//   s   = 0.5*tanh(0.5(a x + b)) + 0.5        (sigmoid)
//   sp  = 0.25(1-th^2) = s(1-s)
//   ssp = s*sp
//   spp = sp(1-2s) = sp - 2*ssp               (needed only for the x^2 term)
//   q   = sp^2 + s*spp = sp*(3sp - s)
// Exact identities used at finalize: Σs^2 = Σs - Σsp; Σspp = Σsp - 2Σssp;
// Σx·spp = Σx·sp - 2Σx·ssp.
// acc: 0:s 1:sp 2:ssp 3:q 4:x*sp 5:x^2*spp 6:x*ssp 7:x*q 8:x^2*q
__device__ __forceinline__ void accum_pair(v2f acc[NACC], v2f xv, v2f A2, v2f B2,
                                           v2f HALF, v2f QTR, v2f NQTR,
                                           v2f THREE, v2f NEG2) {
    v2f hu = pk_fma(A2, xv, B2);       // 0.5*(a*x+b)
    v2f th;
    th.x = fast_tanhf(hu.x);
    th.y = fast_tanhf(hu.y);
    v2f s   = pk_fma(HALF, th, HALF);
    v2f th2 = pk_mul(th, th);
    v2f sp  = pk_fma(NQTR, th2, QTR);
    v2f ssp = pk_mul(s, sp);
    v2f spp = pk_fma(NEG2, ssp, sp);   // sp - 2*ssp
    v2f w   = pk_fma_nc(THREE, sp, s); // 3*sp - s
    v2f q   = pk_mul(sp, w);
    v2f x2  = pk_mul(xv, xv);
    pk_acc_add(acc[0], s);
    pk_acc_add(acc[1], sp);
    pk_acc_add(acc[2], ssp);
    pk_acc_add(acc[3], q);
    pk_acc_fma(acc[4], sp,  xv);
    pk_acc_fma(acc[5], spp, x2);
    pk_acc_fma(acc[6], ssp, xv);
    pk_acc_fma(acc[7], q,   xv);
    pk_acc_fma(acc[8], q,   x2);
}

// scalar version, used only for the N%4 remainder (same 9 components)
__device__ __forceinline__ void accum_one(float acc[NACC], float xv, float a2, float b2) {
    float hu  = __builtin_fmaf(a2, xv, b2);
    float th  = fast_tanhf(hu);
    float s   = __builtin_fmaf(0.5f, th, 0.5f);
    float th2 = th * th;
    float sp  = __builtin_fmaf(-0.25f, th2, 0.25f);
    float ssp = s * sp;
    float spp = __builtin_fmaf(-2.0f, ssp, sp);
    float w   = __builtin_fmaf(3.0f, sp, -s);
    float q   = sp * w;
    float x2  = xv * xv;
    acc[0] += s;  acc[1] += sp;  acc[2] += ssp;  acc[3] += q;
    acc[4] = __builtin_fmaf(sp,  xv, acc[4]);
    acc[5] = __builtin_fmaf(spp, x2, acc[5]);
    acc[6] = __builtin_fmaf(ssp, xv, acc[6]);
    acc[7] = __builtin_fmaf(q,   xv, acc[7]);
    acc[8] = __builtin_fmaf(q,   x2, acc[8]);
}

// ---------------------------------------------------------------------------
// Kernel 1: streaming reduction. Async global->LDS pipeline, DEPTH stages.
// Each lane consumes only the LDS bytes it itself requested => no barriers,
// only per-wave ASYNCcnt discipline. Exactly one async issue per stage
// (clamped dummies at the tail) => constant s_wait_asynccnt DEPTH-1.
// ---------------------------------------------------------------------------
__global__ __launch_bounds__(TPB) void sig_moment_reduce(
    const float* __restrict__ x, const float* __restrict__ a_p,
    const float* __restrict__ b_p, float* __restrict__ partials,
    long long M4, int nIter, int nMain8, int rem)
{
    __shared__ float4 buf[DEPTH][TPB];   // 32 KB staging
    const int tid = threadIdx.x;

    const float av = a_p[0], bv = b_p[0];
    const float a2 = 0.5f * av, b2 = 0.5f * bv;
    const v2f A2 = {a2, a2}, B2 = {b2, b2};
    const v2f HALF = {0.5f, 0.5f}, QTR = {0.25f, 0.25f}, NQTR = {-0.25f, -0.25f};
    const v2f THREE = {3.0f, 3.0f}, NEG2 = {-2.0f, -2.0f};

    v2f acc[NACC];
#pragma unroll
    for (int k = 0; k < NACC; ++k) acc[k] = (v2f){0.0f, 0.0f};

    if (M4 > 0) {
        const unsigned long long srcBase = (unsigned long long)(uintptr_t)x;
        // LDS byte address of this lane's slot in stage-buffer 0.
        // (flat shared-aperture addresses carry the LDS offset in the low 32 bits)
        const uint32_t ldsBase  = (uint32_t)(uintptr_t)&buf[0][tid];
        const long long stride  = (long long)RBLOCKS * TPB;     // float4s per sweep
        const uint32_t vstep    = (uint32_t)(RBLOCKS * TPB * 16);
        const long long gBase   = (long long)blockIdx.x * TPB + tid;

        // ---- prologue: fill the pipeline, clamped (exactly 1 issue per stage)
        {
            long long gI = gBase;
#pragma unroll
            for (int p = 0; p < DEPTH; ++p) {
                long long gcl = (gI < M4) ? gI : (M4 - 1);
                uint32_t voff = (uint32_t)(gcl << 4);
                uint32_t lds  = ldsBase + (uint32_t)(p * TPB * 16);
                asm volatile("global_load_async_to_lds_b128 %0, %1, %2"
                             :: "v"(lds), "v"(voff), "s"(srcBase) : "memory");
                gI += stride;
            }
        }

        // ---- main loop: stages [0, nMain8) -- compute AND issue are clamp-free
        uint32_t voffI = (uint32_t)(gBase << 4) + (uint32_t)DEPTH * vstep;
        for (int st = 0; st < nMain8; st += DEPTH) {
#pragma unroll
            for (int p = 0; p < DEPTH; ++p) {
                asm volatile("s_wait_asynccnt 7" ::: "memory");
                float4 v = buf[p][tid];                 // slot (st+p)&7 == p
                accum_pair(acc, (v2f){v.x, v.y}, A2, B2, HALF, QTR, NQTR, THREE, NEG2);
                accum_pair(acc, (v2f){v.z, v.w}, A2, B2, HALF, QTR, NQTR, THREE, NEG2);
                uint32_t lds = ldsBase + (uint32_t)(p * TPB * 16);
                asm volatile("global_load_async_to_lds_b128 %0, %1, %2"
                             :: "v"(lds), "v"(voffI), "s"(srcBase) : "memory");
                voffI += vstep;
            }
        }

        // ---- generic tail: stages [nMain8, nIter), guarded compute + clamped issue
        long long gC = gBase + (long long)nMain8 * stride;
        long long gI = gBase + (long long)(nMain8 + DEPTH) * stride;
        for (int s = nMain8; s < nIter; ++s) {
            asm volatile("s_wait_asynccnt 7" ::: "memory");
            float4 v = buf[s & (DEPTH - 1)][tid];
            if (gC < M4) {
                accum_pair(acc, (v2f){v.x, v.y}, A2, B2, HALF, QTR, NQTR, THREE, NEG2);
                accum_pair(acc, (v2f){v.z, v.w}, A2, B2, HALF, QTR, NQTR, THREE, NEG2);
            }
            long long gcl = (gI < M4) ? gI : (M4 - 1);
            uint32_t voff = (uint32_t)(gcl << 4);
            uint32_t lds  = ldsBase + (uint32_t)((s & (DEPTH - 1)) * TPB * 16);
            asm volatile("global_load_async_to_lds_b128 %0, %1, %2"
                         :: "v"(lds), "v"(voff), "s"(srcBase) : "memory");
            gI += stride;
            gC += stride;
        }
    }

    // collapse pair lanes
    float accS[NACC];
#pragma unroll
    for (int k = 0; k < NACC; ++k) accS[k] = acc[k].x + acc[k].y;

    // scalar remainder (N % 4), handled once by block 0
    if (blockIdx.x == 0 && tid < rem)
        accum_one(accS, x[4 * M4 + tid], a2, b2);

    // ---- wave32 butterfly reduction
#pragma unroll
    for (int off = 16; off > 0; off >>= 1) {
#pragma unroll
        for (int k = 0; k < NACC; ++k)
            accS[k] += __shfl_down(accS[k], (unsigned)off, 32);
    }

    __shared__ float red[TPB / 32][NACC];
    const int wave = tid >> 5, lane = tid & 31;
    if (lane == 0) {
#pragma unroll
        for (int k = 0; k < NACC; ++k) red[wave][k] = accS[k];
    }
    __syncthreads();
    if (tid < NACC) {
        float s = 0.0f;
#pragma unroll
        for (int w = 0; w < TPB / 32; ++w) s += red[w][tid];
        partials[(long long)blockIdx.x * NACC + tid] = s;
    }
}

// ---------------------------------------------------------------------------
// Kernel 2: sum block partials (double precision) + scalar Newton step.
// 9 waves: wave w reduces component w across all blocks.
// P: 0:Σs 1:Σsp 2:Σssp 3:Σq 4:Σx·sp 5:Σx²·spp 6:Σx·ssp 7:Σx·q 8:Σx²·q
// Derived: Σs² = P0-P1;  Σspp = P1-2P2;  Σx·spp = P4-2P6.
// ---------------------------------------------------------------------------
__global__ __launch_bounds__(NACC * 32) void newton_finalize(
    const float* __restrict__ partials, const float* __restrict__ mean_p,
    const float* __restrict__ var_p, float* __restrict__ out,
    int nPart, long long N)
{
    __shared__ double P[NACC];
    const int wave = threadIdx.x >> 5, lane = threadIdx.x & 31;

    double acc = 0.0;
    for (int bidx = lane; bidx < nPart; bidx += 32)
        acc += (double)partials[(long long)bidx * NACC + wave];
#pragma unroll
    for (int off = 16; off > 0; off >>= 1)
        acc += __shfl_down(acc, (unsigned)off, 32);
    if (lane == 0) P[wave] = acc;
    __syncthreads();

    if (threadIdx.x == 0) {
        const double inv  = 1.0 / (double)N;
        const double mean = (double)mean_p[0];
        const double var  = (double)var_p[0];

        double fm       = P[0] * inv;
        double em       = fm - mean;
        double dem_da   = P[4] * inv;                     // E[x*sp]
        double dem_db   = P[1] * inv;                     // E[sp]
        double d2em_da2 = P[5] * inv;                     // E[x^2*spp]
        double d2em_dab = (P[4] - 2.0 * P[6]) * inv;      // E[x*spp]
        double d2em_db2 = (P[1] - 2.0 * P[2]) * inv;      // E[spp]

        double ev       = (P[0] - P[1]) * inv - fm * fm - var;   // E[f^2]=Σs-Σsp
        double dev_da   = 2.0 * (P[6] * inv - fm * dem_da);      // E[f*df_da]=E[x*ssp]
        double dev_db   = 2.0 * (P[2] * inv - fm * dem_db);      // E[f*df_db]=E[ssp]
        double d2ev_da2 = 2.0 * (P[8] * inv - dem_da * dem_da - fm * d2em_da2);
        double d2ev_dab = 2.0 * (P[7] * inv - dem_da * dem_db - fm * d2em_dab);
        double d2ev_db2 = 2.0 * (P[3] * inv - dem_db * dem_db - fm * d2em_db2);

        double dl_da   = 2.0 * (em * dem_da + ev * dev_da);
        double dl_db   = 2.0 * (em * dem_db + ev * dev_db);
        double d2l_da2 = 2.0 * (dem_da * dem_da + em * d2em_da2 + dev_da * dev_da + ev * d2ev_da2);
        double d2l_dab = 2.0 * (dem_da * dem_db + em * d2em_dab + dev_da * dev_db + ev * d2ev_dab);
        double d2l_db2 = 2.0 * (dem_db * dem_db + em * d2em_db2 + dev_db * dev_db + ev * d2ev_db2);

        double den = d2l_da2 * d2l_db2 - d2l_dab * d2l_dab;
        out[0] = (float)((dl_da * d2l_db2 - dl_db * d2l_dab) / den);
        out[1] = (float)((dl_db * d2l_da2 - dl_da * d2l_dab) / den);
    }
}

// ---------------------------------------------------------------------------
extern "C" void kernel_launch(void* const* d_in, const int* in_sizes, int n_in,
                              void* d_out, int out_size, void* d_ws, size_t ws_size,
                              hipStream_t stream) {
    const float* x    = (const float*)d_in[0];
    const float* a    = (const float*)d_in[1];
    const float* b    = (const float*)d_in[2];
    const float* mean = (const float*)d_in[3];
    const float* var  = (const float*)d_in[4];

    const long long N  = (long long)in_sizes[0];
    const long long M4 = N >> 2;
    const int rem      = (int)(N & 3);
    const long long perSweep = (long long)RBLOCKS * TPB;
    const int nIter    = (int)((M4 + perSweep - 1) / perSweep);
    const int nFull    = (int)(M4 / perSweep);          // stages valid for every lane
    int nMain = (nFull > DEPTH) ? (nFull - DEPTH) : 0;  // compute+issue both clamp-free
    const int nMain8   = (nMain / DEPTH) * DEPTH;       // multiple of DEPTH

    float* partials = (float*)d_ws;   // RBLOCKS * NACC floats = 36 KB

    sig_moment_reduce<<<RBLOCKS, TPB, 0, stream>>>(x, a, b, partials, M4,
                                                   nIter, nMain8, rem);
    newton_finalize<<<1, NACC * 32, 0, stream>>>(partials, mean, var, (float*)d_out,
                                                 RBLOCKS, N);
}